// AT3Batched_26053271617759
// MI455X (gfx1250) — compile-verified
//
#include <hip/hip_runtime.h>
#include <hip/hip_bf16.h>

typedef __attribute__((ext_vector_type(2))) float v2f;
typedef __attribute__((ext_vector_type(8))) float v8f;

#define BATCH   32
#define NNODE   18
#define PER     2048
#define HD      64
#define SPLIT   8
#define PPB     (PER / SPLIT)     // 256 periods per block
#define CHUNK   128
#define NCHUNK  (PPB / CHUNK)     // 2
#define NC      (NNODE * HD)      // 1152

// workspace offsets (in floats)
#define OFS_A      0              // 324
#define OFS_PROBS  512            // 2048
#define OFS_UZ     2560           // 64
#define OFS_CZ     2624           // 64
#define OFS_UH     2688           // 64
#define OFS_CH     2752           // 64
#define OFS_PART   4096           // BATCH*SPLIT*NC = 294912
#define OFS_H      300032         // BATCH*NC = 36864

// fast hardware transcendentals (guarded: fall back to libm device funcs)
#if __has_builtin(__builtin_amdgcn_rcpf)
#define FAST_RCP(x) __builtin_amdgcn_rcpf(x)
#else
#define FAST_RCP(x) (1.0f / (x))
#endif
#if __has_builtin(__builtin_amdgcn_tanhf)
#define FAST_TANH(x) __builtin_amdgcn_tanhf(x)
#elif __has_builtin(__builtin_amdgcn_tanh_f32)
#define FAST_TANH(x) __builtin_amdgcn_tanh_f32(x)
#else
#define FAST_TANH(x) tanhf(x)
#endif

// ---------------------------------------------------------------------------
// Kernel 1: adjacency, softmax(attention), fused gate vectors. 1 block x 256.
// Deterministic: per-output-entry edge scans, fixed-order tree reductions.
// ---------------------------------------------------------------------------
__global__ void at3_setup_kernel(const int* __restrict__ ei, int E,
                                 const float* __restrict__ att,
                                 const float* __restrict__ czw, const float* __restrict__ czb,
                                 const float* __restrict__ chw, const float* __restrict__ chb,
                                 const float* __restrict__ lzw, const float* __restrict__ lzb,
                                 const float* __restrict__ lhw, const float* __restrict__ lhb,
                                 float* __restrict__ ws) {
    __shared__ float dinvS[NNODE];
    __shared__ float red[256];
    const int tid = threadIdx.x;

    // degree (self loop + incoming edges) and dinv, deterministic scan
    if (tid < NNODE) {
        float dg = 1.0f;
        for (int e = 0; e < E; ++e) if (ei[E + e] == tid) dg += 1.0f;
        dinvS[tid] = rsqrtf(fmaxf(dg, 1e-12f));
    }
    __syncthreads();

    // A[m][n] = dinv[n]*dinv[m] * (#edges n->m)  + self-loop dinv[m]^2
    float* wsA = ws + OFS_A;
    for (int i = tid; i < NNODE * NNODE; i += 256) {
        const int m = i / NNODE, n = i % NNODE;
        float a = (m == n) ? dinvS[m] * dinvS[m] : 0.0f;
        float w = dinvS[n] * dinvS[m];
        for (int e = 0; e < E; ++e)
            if (ei[e] == n && ei[E + e] == m) a += w;
        wsA[i] = a;
    }

    // softmax over attention (P = 2048)
    float lmax = -3.4e38f;
    for (int i = tid; i < PER; i += 256) lmax = fmaxf(lmax, att[i]);
    red[tid] = lmax; __syncthreads();
    for (int st = 128; st > 0; st >>= 1) {
        if (tid < st) red[tid] = fmaxf(red[tid], red[tid + st]);
        __syncthreads();
    }
    const float mx = red[0];
    __syncthreads();
    float lsum = 0.0f;
    for (int i = tid; i < PER; i += 256) lsum += expf(att[i] - mx);
    red[tid] = lsum; __syncthreads();
    for (int st = 128; st > 0; st >>= 1) {
        if (tid < st) red[tid] += red[tid + st];
        __syncthreads();
    }
    const float inv = 1.0f / red[0];
    float* probs = ws + OFS_PROBS;
    for (int i = tid; i < PER; i += 256) probs[i] = expf(att[i] - mx) * inv;

    // uz[d] = sum_c conv_z_w[c] * lin_z_w[d,c]   (lin_z_w is (64,128) row-major)
    // cz[d] = sum_c conv_z_b[c] * lin_z_w[d,c] + lin_z_b[d];   same for h
    if (tid < HD) {
        const int d = tid;
        float suz = 0.f, scz = 0.f, suh = 0.f, sch = 0.f;
        for (int c = 0; c < HD; ++c) {
            const float wz = lzw[d * 2 * HD + c];
            const float wh = lhw[d * 2 * HD + c];
            suz = fmaf(czw[c], wz, suz);
            scz = fmaf(czb[c], wz, scz);
            suh = fmaf(chw[c], wh, suh);
            sch = fmaf(chb[c], wh, sch);
        }
        ws[OFS_UZ + d] = suz;
        ws[OFS_CZ + d] = scz + lzb[d];
        ws[OFS_UH + d] = suh;
        ws[OFS_CH + d] = sch + lhb[d];
    }
}

// ---------------------------------------------------------------------------
// Kernel 2: main. grid = BATCH*SPLIT blocks, 256 threads (8 waves).
// Per 128-period chunk: Y = A @ X via v_wmma_f32_16x16x4_f32 (full f32),
// then fused (1-sigmoid)*tanh accumulation weighted by softmax probs.
// ---------------------------------------------------------------------------
__global__ void __launch_bounds__(256)
at3_main_kernel(const float* __restrict__ x, const float* __restrict__ ws_in,
                float* __restrict__ ws_out) {
    __shared__ float Apad[32][20];    // zero-padded 18x18 -> 32x20
    __shared__ float xs[20][CHUNK];   // zero-padded K rows
    __shared__ float ys[32][CHUNK];   // rows 18..31 are exact zeros (A pad)
    __shared__ float uzL[HD], czL[HD], uhL[HD], chL[HD];
    __shared__ float prL[CHUNK];

    const int tid  = threadIdx.x;
    const int lane = tid & 31;
    const int wave = tid >> 5;        // 0..7 == N-tile index
    const int lm   = lane & 15;
    const int hh   = lane >> 4;       // lane half

    const int b = blockIdx.x / SPLIT;
    const int s = blockIdx.x % SPLIT;
    const int p0 = s * PPB;

    const float* Ag = ws_in + OFS_A;
    for (int i = tid; i < 32 * 20; i += 256) {
        const int m = i / 20, k = i % 20;
        Apad[m][k] = (m < NNODE && k < NNODE) ? Ag[m * NNODE + k] : 0.0f;
    }
    if (tid < HD) {
        uzL[tid] = ws_in[OFS_UZ + tid];
        czL[tid] = ws_in[OFS_CZ + tid];
        uhL[tid] = ws_in[OFS_UH + tid];
        chL[tid] = ws_in[OFS_CH + tid];
    }
    __syncthreads();

    const int d = tid & 63;           // hidden channel owned by this thread
    const int g = tid >> 6;           // node group (m = g, g+4, g+8, ...)
    float acc[5] = {0.f, 0.f, 0.f, 0.f, 0.f};

    for (int ch = 0; ch < NCHUNK; ++ch) {
        const int pc = p0 + ch * CHUNK;
        // stage X panel (18 x 128, coalesced rows), zero pad rows 18..19
        for (int i = tid; i < 20 * CHUNK; i += 256) {
            const int n = i >> 7, col = i & (CHUNK - 1);
            xs[n][col] = (n < NNODE)
                ? x[((size_t)b * NNODE + n) * PER + pc + col] : 0.0f;
        }
        if (tid < CHUNK) prL[tid] = ws_in[OFS_PROBS + pc + tid];
        __syncthreads();

        // Y(32x128) = Apad(32x20) @ xs(20x128); wave handles N-tile `wave`
#pragma unroll
        for (int mt = 0; mt < 2; ++mt) {
            v8f c = {0.f, 0.f, 0.f, 0.f, 0.f, 0.f, 0.f, 0.f};
#pragma unroll
            for (int ks = 0; ks < 5; ++ks) {
                const int k0 = ks * 4;
                v2f a, bf;
                // A 16x4 layout: lanes 0-15 -> K0,K1; lanes 16-31 -> K2,K3
                a.x = Apad[mt * 16 + lm][k0 + 2 * hh + 0];
                a.y = Apad[mt * 16 + lm][k0 + 2 * hh + 1];
                // B 4x16 layout mirrors A with N across lanes
                bf.x = xs[k0 + 2 * hh + 0][wave * 16 + lm];
                bf.y = xs[k0 + 2 * hh + 1][wave * 16 + lm];
                c = __builtin_amdgcn_wmma_f32_16x16x4_f32(
                        false, a, false, bf, (short)0, c, false, false);
            }
            // D layout: VGPR r -> row r (lanes 0-15) / r+8 (lanes 16-31)
#pragma unroll
            for (int r = 0; r < 8; ++r) {
                const int m = mt * 16 + 8 * hh + r;   // 0..31, no guard needed
                ys[m][wave * 16 + lm] = c[r];
            }
        }
        __syncthreads();

        // fused gate: hacc[m][d] += pr * (1 - sigmoid(y*uz+cz)) * tanh(y*uh+ch)
        const float uzd = uzL[d], czd = czL[d], uhd = uhL[d], chd = chL[d];
        for (int pl = 0; pl < CHUNK; ++pl) {
            const float pr = prL[pl];
#pragma unroll
            for (int j = 0; j < 5; ++j) {
                const int m = g + 4 * j;              // wave-uniform guard
                if (m < NNODE) {
                    const float y = ys[m][pl];
                    const float zs = FAST_RCP(1.0f + __expf(-fmaf(y, uzd, czd)));
                    const float ht = FAST_TANH(fmaf(y, uhd, chd));
                    acc[j] = fmaf(pr, (1.0f - zs) * ht, acc[j]);
                }
            }
        }
        __syncthreads();   // before next chunk overwrites xs/ys
    }

    float* part = ws_out + OFS_PART + (size_t)(b * SPLIT + s) * NC;
#pragma unroll
    for (int j = 0; j < 5; ++j) {
        const int m = g + 4 * j;
        if (m < NNODE) part[m * HD + d] = acc[j];
    }
}

// ---------------------------------------------------------------------------
// Kernel 3: reduce split partials + BatchNorm over (B,C) per node + leaky.
// grid = NNODE blocks x 256 threads. Deterministic tree reduction.
// ---------------------------------------------------------------------------
__global__ void at3_bn_kernel(const float* __restrict__ ws_in,
                              const float* __restrict__ gamma,
                              const float* __restrict__ beta,
                              float* __restrict__ ws_out) {
    __shared__ float rs[256], rq[256];
    const int n = blockIdx.x;
    const int tid = threadIdx.x;
    const float* part = ws_in + OFS_PART;

    float vals[8];
    float lsum = 0.f, lsq = 0.f;
#pragma unroll
    for (int r = 0; r < 8; ++r) {
        const int idx = tid + r * 256;     // (b,c) pair, 32*64 = 2048 total
        const int bb = idx >> 6, c = idx & 63;
        float v = 0.f;
        for (int s2 = 0; s2 < SPLIT; ++s2)
            v += part[(size_t)(bb * SPLIT + s2) * NC + n * HD + c];
        vals[r] = v; lsum += v; lsq = fmaf(v, v, lsq);
    }
    rs[tid] = lsum; rq[tid] = lsq; __syncthreads();
    for (int st = 128; st > 0; st >>= 1) {
        if (tid < st) { rs[tid] += rs[tid + st]; rq[tid] += rq[tid + st]; }
        __syncthreads();
    }
    const float mean = rs[0] * (1.0f / 2048.0f);
    const float var  = rq[0] * (1.0f / 2048.0f) - mean * mean;
    const float scale = rsqrtf(var + 1e-5f) * gamma[n];
    const float bta = beta[n];
    float* hout = ws_out + OFS_H;
#pragma unroll
    for (int r = 0; r < 8; ++r) {
        const int idx = tid + r * 256;
        const int bb = idx >> 6, c = idx & 63;
        float hv = fmaf(vals[r] - mean, scale, bta);
        hv = (hv >= 0.f) ? hv : 0.01f * hv;
        hout[(size_t)bb * NC + n * HD + c] = hv;
    }
}

// ---------------------------------------------------------------------------
// Kernel 4: fc1 -> leaky -> fc2 -> leaky -> fc3. 1 block x 256 threads.
// ---------------------------------------------------------------------------
__global__ void at3_fc_kernel(const float* __restrict__ ws_in,
                              const float* __restrict__ fc1w, const float* __restrict__ fc1b,
                              const float* __restrict__ fc2w, const float* __restrict__ fc2b,
                              const float* __restrict__ fc3w, const float* __restrict__ fc3b,
                              float* __restrict__ out) {
    __shared__ float a1[BATCH * 64];
    __shared__ float a2[BATCH * 32];
    const int tid = threadIdx.x;
    const float* h = ws_in + OFS_H;

    for (int o = tid; o < BATCH * 64; o += 256) {
        const int bb = o >> 6, j = o & 63;
        const float* hrow = h + (size_t)bb * NC;
        const float* wrow = fc1w + (size_t)j * NC;
        float sum = fc1b[j];
        for (int k = 0; k < NC; ++k) sum = fmaf(hrow[k], wrow[k], sum);
        a1[o] = (sum >= 0.f) ? sum : 0.01f * sum;
    }
    __syncthreads();
    for (int o = tid; o < BATCH * 32; o += 256) {
        const int bb = o >> 5, j = o & 31;
        float sum = fc2b[j];
#pragma unroll
        for (int k = 0; k < 64; ++k) sum = fmaf(a1[bb * 64 + k], fc2w[j * 64 + k], sum);
        a2[o] = (sum >= 0.f) ? sum : 0.01f * sum;
    }
    __syncthreads();
    if (tid < BATCH) {
        float sum = fc3b[0];
#pragma unroll
        for (int k = 0; k < 32; ++k) sum = fmaf(a2[tid * 32 + k], fc3w[k], sum);
        out[tid] = sum;
    }
}

// ---------------------------------------------------------------------------
extern "C" void kernel_launch(void* const* d_in, const int* in_sizes, int n_in,
                              void* d_out, int out_size, void* d_ws, size_t ws_size,
                              hipStream_t stream) {
    const float* x      = (const float*)d_in[0];
    const int*   ei     = (const int*)  d_in[1];
    const float* att    = (const float*)d_in[2];
    const float* czw    = (const float*)d_in[3];
    const float* czb    = (const float*)d_in[4];
    const float* chw    = (const float*)d_in[5];
    const float* chb    = (const float*)d_in[6];
    const float* lzw    = (const float*)d_in[7];
    const float* lzb    = (const float*)d_in[8];
    const float* lhw    = (const float*)d_in[9];
    const float* lhb    = (const float*)d_in[10];
    const float* gamma  = (const float*)d_in[11];
    const float* beta   = (const float*)d_in[12];
    const float* fc1w   = (const float*)d_in[13];
    const float* fc1b   = (const float*)d_in[14];
    const float* fc2w   = (const float*)d_in[15];
    const float* fc2b   = (const float*)d_in[16];
    const float* fc3w   = (const float*)d_in[17];
    const float* fc3b   = (const float*)d_in[18];
    float* out = (float*)d_out;
    float* ws  = (float*)d_ws;
    const int E = in_sizes[1] / 2;

    at3_setup_kernel<<<1, 256, 0, stream>>>(ei, E, att, czw, czb, chw, chb,
                                            lzw, lzb, lhw, lhb, ws);
    at3_main_kernel<<<BATCH * SPLIT, 256, 0, stream>>>(x, ws, ws);
    at3_bn_kernel<<<NNODE, 256, 0, stream>>>(ws, gamma, beta, ws);
    at3_fc_kernel<<<1, 256, 0, stream>>>(ws, fc1w, fc1b, fc2w, fc2b, fc3w, fc3b, out);
}